// GeneralSampleEdgeConv_17008070492326
// MI455X (gfx1250) — compile-verified
//
#include <hip/hip_runtime.h>

typedef float v2f __attribute__((ext_vector_type(2)));
typedef float v8f __attribute__((ext_vector_type(8)));

#define N_NODES   100000
#define N_EDGES   1600000
#define DIM       128          // dim_out, dim_in, edge_dim
#define KTOT      256          // dim_in + edge_dim
#define TILE_E    128          // edges per workgroup tile
#define N_TILES   (N_EDGES / TILE_E)   // 12500 exactly
#define NUM_WG    512

// ---- LDS layout (bytes) ----
// B (W transposed, k-pair major): row per kp=k/2, each row 128 lane-pairs (2 floats)
// padded to 288 words so lanes 0-15 (kp) and 16-31 (kp+1) hit disjoint bank sets.
#define B_PITCH_B (288 * 4)                 // 1152 B per kp row
#define B_BYTES   (128 * B_PITCH_B)         // 147456
// A (edge rows, row-major k): pitch 260 words -> bank stride 4 across rows.
#define A_PITCH_B (260 * 4)                 // 1040 B per edge row (16B aligned)
#define A_OFF     B_BYTES
#define A_BYTES   (TILE_E * A_PITCH_B)      // 133120
#define DST_OFF   (A_OFF + A_BYTES)
#define MASK_OFF  (DST_OFF + TILE_E * 4)
#define LDS_BYTES (MASK_OFF + TILE_E * 4)   // 281600 <= 320KB

extern "C" __global__ void gnn_zero_out(float4* __restrict__ out, int n4) {
  int i = blockIdx.x * blockDim.x + threadIdx.x;
  if (i < n4) out[i] = make_float4(0.f, 0.f, 0.f, 0.f);
}

extern "C" __global__ __launch_bounds__(256)
void gnn_edge_wmma(const float* __restrict__ x,
                   const long long* __restrict__ edge_index,
                   const float* __restrict__ edge_attr,
                   const unsigned char* __restrict__ edge_mask,
                   const float* __restrict__ W,
                   float* __restrict__ out)
{
  extern __shared__ char lds[];

  const int t    = threadIdx.x;
  const int lane = t & 31;
  const int wave = t >> 5;
  const int mgrp = wave & 3;   // edge group: rows mgrp*32 .. +31
  const int ngrp = wave >> 2;  // col group:  cols ngrp*64 .. +63
  const int lhi  = lane >> 4;  // 0: lanes 0-15, 1: lanes 16-31
  const int llo  = lane & 15;

  // ---------------- stage W into LDS (once per workgroup) ----------------
  // layout: element (k, n) at byte (k/2)*B_PITCH_B + n*8 + (k&1)*4
  {
    const int n  = t >> 1;        // 0..127 output column
    const int kh = t & 1;         // which half of k-range
    const float4* wrow = (const float4*)(W + n * KTOT + kh * 128);
#pragma unroll
    for (int j = 0; j < 32; ++j) {
      float4 w = wrow[j];
      int k  = kh * 128 + j * 4;
      int kp = k >> 1;
      *(float2*)(lds + (kp    ) * B_PITCH_B + n * 8) = make_float2(w.x, w.y);
      *(float2*)(lds + (kp + 1) * B_PITCH_B + n * 8) = make_float2(w.z, w.w);
    }
  }

  // per-lane fragment base addresses (byte offsets into lds)
  const char* aBase0 = lds + A_OFF + (mgrp * 32 + llo) * A_PITCH_B + lhi * 8;
  const char* aBase1 = aBase0 + 16 * A_PITCH_B;
  const char* bBase  = lds + lhi * B_PITCH_B + (ngrp * 64 + llo) * 8;

  const int* dstl   = (const int*)(lds + DST_OFF);
  const float* mskl = (const float*)(lds + MASK_OFF);

  // ---------------- register double-buffer for A-panel staging ----------------
  const int e = t >> 1;     // local edge 0..127 staged by this thread
  const int h = t & 1;      // 0: x[src] half (+ dst/mask), 1: edge_attr half
  float4 regbuf[8];
  int    nxt_dst  = 0;
  float  nxt_mask = 0.f;

  auto stage_load = [&](int tile) {
    const int ebase = tile * TILE_E;
    const float* srcp;
    if (h == 0) {
      long long s = edge_index[ebase + e];                    // src node
      srcp = x + (int)s * DIM;
      nxt_dst  = (int)edge_index[N_EDGES + ebase + e];
      nxt_mask = edge_mask[ebase + e] ? 1.0f : 0.0f;
    } else {
      srcp = edge_attr + (long long)(ebase + e) * DIM;
    }
#pragma unroll
    for (int j = 0; j < 8; ++j) regbuf[j] = ((const float4*)srcp)[j];
  };

  auto stage_store = [&]() {
    char* drow = lds + A_OFF + e * A_PITCH_B + h * 512;
#pragma unroll
    for (int j = 0; j < 8; ++j) *(float4*)(drow + j * 16) = regbuf[j];
    if (h == 0) {
      ((int*)(lds + DST_OFF))[e]    = nxt_dst;
      ((float*)(lds + MASK_OFF))[e] = nxt_mask;
    }
  };

  int tile = blockIdx.x;
  if (tile < N_TILES) stage_load(tile);          // prime the pipeline

  for (; tile < N_TILES; tile += gridDim.x) {
    __syncthreads();   // previous tile's LDS readers done before overwriting A
    stage_store();

    // kick off next tile's global gather now; its waits land AFTER the WMMA loop
    const int ntile = tile + gridDim.x;          // block-uniform branch
    if (ntile < N_TILES) stage_load(ntile);

    __syncthreads();

    // ---------------- compute: 32 edges x 64 cols per wave, exact fp32 WMMA ----
    v8f acc[2][4];
#pragma unroll
    for (int mt = 0; mt < 2; ++mt)
#pragma unroll
      for (int nt = 0; nt < 4; ++nt)
        acc[mt][nt] = (v8f){0.f, 0.f, 0.f, 0.f, 0.f, 0.f, 0.f, 0.f};

    const char* pa0 = aBase0;
    const char* pa1 = aBase1;
    const char* pb  = bBase;
#pragma unroll 4
    for (int ks = 0; ks < 64; ++ks) {   // k0 = 4*ks
      v2f a0 = *(const v2f*)pa0;
      v2f a1 = *(const v2f*)pa1;
      v2f b0 = *(const v2f*)(pb);
      v2f b1 = *(const v2f*)(pb + 128);
      v2f b2 = *(const v2f*)(pb + 256);
      v2f b3 = *(const v2f*)(pb + 384);
      acc[0][0] = __builtin_amdgcn_wmma_f32_16x16x4_f32(false, a0, false, b0, (short)0, acc[0][0], false, false);
      acc[0][1] = __builtin_amdgcn_wmma_f32_16x16x4_f32(false, a0, false, b1, (short)0, acc[0][1], false, false);
      acc[0][2] = __builtin_amdgcn_wmma_f32_16x16x4_f32(false, a0, false, b2, (short)0, acc[0][2], false, false);
      acc[0][3] = __builtin_amdgcn_wmma_f32_16x16x4_f32(false, a0, false, b3, (short)0, acc[0][3], false, false);
      acc[1][0] = __builtin_amdgcn_wmma_f32_16x16x4_f32(false, a1, false, b0, (short)0, acc[1][0], false, false);
      acc[1][1] = __builtin_amdgcn_wmma_f32_16x16x4_f32(false, a1, false, b1, (short)0, acc[1][1], false, false);
      acc[1][2] = __builtin_amdgcn_wmma_f32_16x16x4_f32(false, a1, false, b2, (short)0, acc[1][2], false, false);
      acc[1][3] = __builtin_amdgcn_wmma_f32_16x16x4_f32(false, a1, false, b3, (short)0, acc[1][3], false, false);
      pa0 += 16; pa1 += 16; pb += 2 * B_PITCH_B;
    }

    // ---------------- epilogue: mask + scatter-add (atomics resolve in L2) ----
    // C element v of a 16x16 tile: M = v + 8*lhi, N = llo
#pragma unroll
    for (int mt = 0; mt < 2; ++mt) {
#pragma unroll
      for (int v = 0; v < 8; ++v) {
        const int r = mgrp * 32 + mt * 16 + lhi * 8 + v;   // local edge row
        const float m = mskl[r];
        if (m != 0.0f) {
          float* orow = out + dstl[r] * DIM + ngrp * 64 + llo;
          unsafeAtomicAdd(orow +  0, acc[mt][0][v]);
          unsafeAtomicAdd(orow + 16, acc[mt][1][v]);
          unsafeAtomicAdd(orow + 32, acc[mt][2][v]);
          unsafeAtomicAdd(orow + 48, acc[mt][3][v]);
        }
      }
    }
  }
}

extern "C" void kernel_launch(void* const* d_in, const int* in_sizes, int n_in,
                              void* d_out, int out_size, void* d_ws, size_t ws_size,
                              hipStream_t stream) {
  const float*         x  = (const float*)d_in[0];
  const long long*     ei = (const long long*)d_in[1];   // [2, E] int64
  const float*         ea = (const float*)d_in[2];
  const unsigned char* em = (const unsigned char*)d_in[3]; // bool
  const float*         W  = (const float*)d_in[4];       // [128, 256]
  float* out = (float*)d_out;

  // out is poisoned by the harness; zero it (same stream => ordered before atomics)
  const int n4 = N_NODES * DIM / 4;
  gnn_zero_out<<<(n4 + 255) / 256, 256, 0, stream>>>((float4*)out, n4);

  // 281,600 B dynamic LDS per workgroup (320 KB available per WGP on CDNA5)
  hipFuncSetAttribute((const void*)gnn_edge_wmma,
                      hipFuncAttributeMaxDynamicSharedMemorySize, LDS_BYTES);
  gnn_edge_wmma<<<NUM_WG, 256, LDS_BYTES, stream>>>(x, ei, ea, em, W, out);
}